// MultiLoraLinear_63883343560843
// MI455X (gfx1250) — compile-verified
//
#include <hip/hip_runtime.h>
#include <stdint.h>

typedef __attribute__((ext_vector_type(2))) float v2f;
typedef __attribute__((ext_vector_type(4))) float v4f;
typedef __attribute__((ext_vector_type(8))) float v8f;

#define B_    8
#define S_    2048
#define IN_   4096
#define OUT_  64
#define WAVES 8
#define ROWS_PER_WAVE  32
#define ROWS_PER_BLOCK (WAVES * ROWS_PER_WAVE)   // 256
#define KSPLIT 8
#define KLEN   (IN_ / KSPLIT)                    // 512 per split

#define KC        32                              // K columns per staged tile
#define NTILES    (KLEN / KC)                     // 16
#define NBUF      3                               // async pipeline depth
#define PITCH_F   36                              // LDS row pitch (dwords): bank-conflict-free
#define PITCH_B   (PITCH_F * 4)                   // 144 bytes
#define TILE_F    (ROWS_PER_WAVE * PITCH_F)       // 1152 floats per tile buffer
#define TILE_B    (TILE_F * 4)                    // 4608 bytes
#define WAVE_LDS_F (NBUF * TILE_F)                // 3456 floats per wave
#define WAVE_LDS_B (WAVE_LDS_F * 4)               // 13824 bytes

// ---------------------------------------------------------------------------
// Pass 0: zero the output (harness poisons d_out; split-K accumulates into it)
// ---------------------------------------------------------------------------
__global__ __launch_bounds__(256) void zero_out_kernel(float* __restrict__ out,
                                                       int n_float4) {
    int i = blockIdx.x * blockDim.x + threadIdx.x;
    if (i < n_float4) {
        ((v4f*)out)[i] = (v4f){};
    }
}

// ---------------------------------------------------------------------------
// Pass 1: split-K WMMA GEMM with async-LDS staging of the x stream.
//   - x tile (32 rows x 32 k floats per wave) staged via
//     global_load_async_to_lds_b128 (coalesced 512B/instr, ASYNCcnt pipelined,
//     3 tiles deep, non-temporal).
//   - A/B frags per CDNA5 ISA 7.12.2; D = A*B + C with V_WMMA_F32_16X16X4_F32.
//   - split-K partials accumulated with hardware f32 atomics.
// ---------------------------------------------------------------------------
__global__ __launch_bounds__(256) void multilora_wmma_f32(
    const float* __restrict__ x,      // [B, S, IN]
    const int*   __restrict__ ids,    // [B]
    const float* __restrict__ w,      // [L, OUT, IN]
    float*       __restrict__ out)    // [B, S, OUT]
{
    __shared__ float lds_x[WAVES * WAVE_LDS_F];   // 108 KB

    const int tid  = threadIdx.x;
    const int wave = tid >> 5;
    const int lane = tid & 31;
    const int half = lane >> 4;   // 0: lanes 0-15, 1: lanes 16-31
    const int l15  = lane & 15;

    const int b    = blockIdx.y;
    const int row0 = blockIdx.x * ROWS_PER_BLOCK + wave * ROWS_PER_WAVE;
    const int kbeg = blockIdx.z * KLEN;
    const int aid  = ids[b];

    // ---- staging addresses (4 rows x 128B per async b128 instruction) ----
    // lane -> row sub-index (lane>>3), 16B segment (lane&7)
    const uint64_t g_lane = (uint64_t)(uintptr_t)(
        x + ((size_t)b * S_ + row0 + (lane >> 3)) * IN_ + kbeg + (lane & 7) * 4);
    const uint32_t l_lane = (uint32_t)(uintptr_t)lds_x
        + (uint32_t)wave * WAVE_LDS_B
        + (uint32_t)(lane >> 3) * PITCH_B
        + (uint32_t)(lane & 7) * 16;

    // B-frag source rows: w[aid, n*16 + l15, :]  (row of W == column of B)
    const float* wr = w + ((size_t)aid * OUT_ + l15) * IN_;

    v8f acc[2][4];
    #pragma unroll
    for (int m = 0; m < 2; ++m)
        #pragma unroll
        for (int n = 0; n < 4; ++n)
            acc[m][n] = (v8f){};

    const int koff = 2 * half;    // per-lane K sub-offset inside a 4-wide K step

    // Issue one tile's 8 async b128 loads (32 rows x 128B -> LDS buffer).
    // s_wait_dscnt 0 first: the buffer being refilled was read by ds_loads.
#define ISSUE_TILE(T, BUF) do {                                               \
        asm volatile("s_wait_dscnt 0" ::: "memory");                          \
        const uint64_t gt = g_lane + (uint64_t)(T) * (KC * 4);                \
        const uint32_t lt = l_lane + (uint32_t)(BUF) * TILE_B;                \
        _Pragma("unroll")                                                     \
        for (int i = 0; i < 8; ++i) {                                         \
            asm volatile(                                                     \
                "global_load_async_to_lds_b128 %0, %1, off th:TH_LOAD_NT"     \
                :: "v"(lt + (uint32_t)i * (4 * PITCH_B)),                     \
                   "v"(gt + (uint64_t)i * (4 * (uint64_t)IN_ * 4))            \
                : "memory");                                                  \
        }                                                                     \
    } while (0)

    // Consume one staged tile: 8 K-steps x (2 A ds_loads + 4 B global loads + 8 WMMA).
#define COMPUTE_TILE(T, BUF) do {                                             \
        const float* lb = lds_x + wave * WAVE_LDS_F + (BUF) * TILE_F;         \
        const int kt = kbeg + (T) * KC;                                       \
        _Pragma("unroll")                                                     \
        for (int kk = 0; kk < KC; kk += 4) {                                  \
            v2f a0 = *(const v2f*)(lb + (l15)      * PITCH_F + kk + koff);    \
            v2f a1 = *(const v2f*)(lb + (16 + l15) * PITCH_F + kk + koff);    \
            const int k0 = kt + kk;                                           \
            v2f bf0 = *(const v2f*)(wr + (size_t)0 * 16 * IN_ + k0 + koff);   \
            v2f bf1 = *(const v2f*)(wr + (size_t)1 * 16 * IN_ + k0 + koff);   \
            v2f bf2 = *(const v2f*)(wr + (size_t)2 * 16 * IN_ + k0 + koff);   \
            v2f bf3 = *(const v2f*)(wr + (size_t)3 * 16 * IN_ + k0 + koff);   \
            acc[0][0] = __builtin_amdgcn_wmma_f32_16x16x4_f32(false, a0, false, bf0, (short)0, acc[0][0], false, false); \
            acc[0][1] = __builtin_amdgcn_wmma_f32_16x16x4_f32(false, a0, false, bf1, (short)0, acc[0][1], false, false); \
            acc[0][2] = __builtin_amdgcn_wmma_f32_16x16x4_f32(false, a0, false, bf2, (short)0, acc[0][2], false, false); \
            acc[0][3] = __builtin_amdgcn_wmma_f32_16x16x4_f32(false, a0, false, bf3, (short)0, acc[0][3], false, false); \
            acc[1][0] = __builtin_amdgcn_wmma_f32_16x16x4_f32(false, a1, false, bf0, (short)0, acc[1][0], false, false); \
            acc[1][1] = __builtin_amdgcn_wmma_f32_16x16x4_f32(false, a1, false, bf1, (short)0, acc[1][1], false, false); \
            acc[1][2] = __builtin_amdgcn_wmma_f32_16x16x4_f32(false, a1, false, bf2, (short)0, acc[1][2], false, false); \
            acc[1][3] = __builtin_amdgcn_wmma_f32_16x16x4_f32(false, a1, false, bf3, (short)0, acc[1][3], false, false); \
        }                                                                     \
    } while (0)

    // ---- 3-deep async pipeline over 16 tiles (per-wave private: no barriers)
    ISSUE_TILE(0, 0);
    ISSUE_TILE(1, 1);
    ISSUE_TILE(2, 2);

    int buf = 0;
    for (int t = 0; t < NTILES - NBUF; ++t) {            // tiles 0..12
        // 3 tiles (24 loads) outstanding; <=16 left => oldest tile landed.
        asm volatile("s_wait_asynccnt 16" ::: "memory");
        COMPUTE_TILE(t, buf);
        ISSUE_TILE(t + NBUF, buf);                        // refill same buffer
        buf = (buf == NBUF - 1) ? 0 : buf + 1;
    }
    // Drain: tiles 13, 14, 15 with exact waits.
    asm volatile("s_wait_asynccnt 16" ::: "memory");
    COMPUTE_TILE(NTILES - 3, buf);
    buf = (buf == NBUF - 1) ? 0 : buf + 1;
    asm volatile("s_wait_asynccnt 8" ::: "memory");
    COMPUTE_TILE(NTILES - 2, buf);
    buf = (buf == NBUF - 1) ? 0 : buf + 1;
    asm volatile("s_wait_asynccnt 0" ::: "memory");
    COMPUTE_TILE(NTILES - 1, buf);

#undef ISSUE_TILE
#undef COMPUTE_TILE

    // C/D layout: VGPR r -> row (r + 8*half), col = l15, per 16-col N tile.
    // Split-K partials -> hardware f32 atomic add (non-returning).
    #pragma unroll
    for (int m = 0; m < 2; ++m) {
        float* obase = out + ((size_t)b * S_ + row0 + m * 16 + 8 * half) * OUT_;
        #pragma unroll
        for (int n = 0; n < 4; ++n) {
            float* op = obase + n * 16 + l15;
            #pragma unroll
            for (int r = 0; r < 8; ++r) {
                unsafeAtomicAdd(op + (size_t)r * OUT_, acc[m][n][r]);
            }
        }
    }
}

extern "C" void kernel_launch(void* const* d_in, const int* in_sizes, int n_in,
                              void* d_out, int out_size, void* d_ws, size_t ws_size,
                              hipStream_t stream) {
    (void)in_sizes; (void)n_in; (void)d_ws; (void)ws_size;
    const float* x   = (const float*)d_in[0];
    const int*   ids = (const int*)  d_in[1];
    const float* w   = (const float*)d_in[2];
    float*       out = (float*)      d_out;

    // Pass 0: zero the 4 MB output (split-K accumulates into it).
    const int n_f4 = out_size / 4;                       // 262144 float4
    zero_out_kernel<<<(n_f4 + 255) / 256, 256, 0, stream>>>(out, n_f4);

    // Pass 1: 512 blocks x 8 waves = 4096 waves; ~48 MB of async loads in flight.
    dim3 grid(S_ / ROWS_PER_BLOCK, B_, KSPLIT);          // (8, 8, 8)
    dim3 block(256);
    multilora_wmma_f32<<<grid, block, 0, stream>>>(x, ids, w, out);
}